// NetM_27943057227873
// MI455X (gfx1250) — compile-verified
//
#include <hip/hip_runtime.h>
#include <math.h>

typedef __attribute__((ext_vector_type(2))) float v2f;
typedef __attribute__((ext_vector_type(8))) float v8f;

// ---------------------------------------------------------------------------
// f32 WMMA: D = A(16x4) x B(4x16) + C, full fp32 (matches reference dtype)
// ---------------------------------------------------------------------------
__device__ inline v8f wmma_f32_16x16x4(v2f a, v2f b, v8f c) {
#if __has_builtin(__builtin_amdgcn_wmma_f32_16x16x4_f32)
  return __builtin_amdgcn_wmma_f32_16x16x4_f32(
      /*neg_a=*/false, a, /*neg_b=*/false, b,
      /*c_mod=*/(short)0, c, /*reuse_a=*/false, /*reuse_b=*/false);
#else
  c[0] += a[0] * b[0] + a[1] * b[1];
  return c;
#endif
}

// ---------------------------------------------------------------------------
// zero scratch (float4 stores)
// ---------------------------------------------------------------------------
__global__ void zero_kernel(float4* __restrict__ p, long long n4) {
  long long stride = (long long)gridDim.x * blockDim.x;
  for (long long i = (long long)blockIdx.x * blockDim.x + threadIdx.x; i < n4; i += stride)
    p[i] = make_float4(0.f, 0.f, 0.f, 0.f);
}

// ---------------------------------------------------------------------------
// degrees: outdeg over row (source), indeg over col (target)
// ---------------------------------------------------------------------------
__global__ void degree_kernel(const int* __restrict__ ei, int E,
                              unsigned* __restrict__ outdeg,
                              unsigned* __restrict__ indeg) {
  int stride = gridDim.x * blockDim.x;
  for (int e = blockIdx.x * blockDim.x + threadIdx.x; e < E; e += stride) {
    atomicAdd(&outdeg[ei[e]], 1u);
    atomicAdd(&indeg[ei[E + e]], 1u);
  }
}

// dis[v] = rsqrt(outdeg+1);  coef[v] = indeg + 1 + dis^2
__global__ void node_prep_kernel(const unsigned* __restrict__ outdeg,
                                 const unsigned* __restrict__ indeg,
                                 float* __restrict__ dis, float* __restrict__ coef,
                                 int N) {
  int stride = gridDim.x * blockDim.x;
  for (int v = blockIdx.x * blockDim.x + threadIdx.x; v < N; v += stride) {
    float d = rsqrtf((float)outdeg[v] + 1.0f);
    dis[v]  = d;
    coef[v] = (float)indeg[v] + 1.0f + d * d;
  }
}

// ---------------------------------------------------------------------------
// WMMA GEMM (compile-time shapes so accumulators live in VGPRs and the ct/i loops
// fully unroll):  C[N,FOUT] = A[N,FIN] * W[FOUT,FIN]^T + bias
// One wave per 16-row stripe; wave holds FOUT/16 v8f accumulators (all cols).
// ---------------------------------------------------------------------------
template <int FIN, int FOUT>
__global__ __launch_bounds__(256) void gemm_bias_wmma(
    const float* __restrict__ A, const float* __restrict__ W,
    const float* __restrict__ bias, float* __restrict__ C, int nStripes) {
  constexpr int NT = FOUT / 16;  // 4 or 8 column tiles
  int wid = (blockIdx.x * blockDim.x + threadIdx.x) >> 5;
  if (wid >= nStripes) return;
  int lane = threadIdx.x & 31;
  int m = lane & 15;      // row within A-frag / col within B-frag / col of C
  int half = lane >> 4;   // k sub-slot selector

  long long rowbase = (long long)wid * 16;
  v8f acc[NT];
#pragma unroll
  for (int ct = 0; ct < NT; ++ct) {
    float bv = bias[ct * 16 + m];
#pragma unroll
    for (int i = 0; i < 8; ++i) acc[ct][i] = bv;
  }

  const float* arow = A + (rowbase + m) * (long long)FIN;
#pragma unroll 4
  for (int k = 0; k < FIN; k += 4) {
    int kk = k + half * 2;
    v2f av = *(const v2f*)(arow + kk);  // {A[m][kk], A[m][kk+1]}
#pragma unroll
    for (int ct = 0; ct < NT; ++ct) {
      // B[k][n] = W[n][k], n = ct*16 + m
      v2f bv = *(const v2f*)(W + (long long)(ct * 16 + m) * FIN + kk);
      acc[ct] = wmma_f32_16x16x4(av, bv, acc[ct]);
    }
  }

  // C layout: VGPR i -> row = i + 8*half, col = lane&15
#pragma unroll
  for (int ct = 0; ct < NT; ++ct)
#pragma unroll
    for (int i = 0; i < 8; ++i)
      C[(rowbase + i + 8 * half) * (long long)FOUT + ct * 16 + m] = acc[ct][i];
}

// generic fallback / tail rows
__global__ void gemm_naive_kernel(const float* __restrict__ A, const float* __restrict__ W,
                                  const float* __restrict__ bias, float* __restrict__ C,
                                  int rowStart, int N, int Fin, int Fout) {
  long long total = (long long)(N - rowStart) * Fout;
  long long stride = (long long)gridDim.x * blockDim.x;
  for (long long i = (long long)blockIdx.x * blockDim.x + threadIdx.x; i < total; i += stride) {
    int r = rowStart + (int)(i / Fout);
    int c = (int)(i % Fout);
    float s = bias[c];
    for (int k = 0; k < Fin; ++k) s += A[(long long)r * Fin + k] * W[(long long)c * Fin + k];
    C[(long long)r * Fout + c] = s;
  }
}

// dispatch helper: WMMA path for the two known shapes, naive otherwise
static void launch_gemm(const float* A, const float* W, const float* bias, float* C,
                        long long N, int Fin, int Fout, hipStream_t stream) {
  int stripes = (int)(N / 16);
  int tail = (int)(N % 16);
  int blocks = (stripes + 7) / 8;
  if (Fin == 128 && Fout == 64) {
    gemm_bias_wmma<128, 64><<<blocks, 256, 0, stream>>>(A, W, bias, C, stripes);
  } else if (Fin == 64 && Fout == 128) {
    gemm_bias_wmma<64, 128><<<blocks, 256, 0, stream>>>(A, W, bias, C, stripes);
  } else {
    gemm_naive_kernel<<<2048, 256, 0, stream>>>(A, W, bias, C, 0, (int)N, Fin, Fout);
    return;
  }
  if (tail)
    gemm_naive_kernel<<<64, 256, 0, stream>>>(A, W, bias, C, (int)(N - tail), (int)N, Fin, Fout);
}

// ---------------------------------------------------------------------------
// edge scatter: acc[col] += dis[row] * h[row]   (one wave per edge)
// ---------------------------------------------------------------------------
__global__ __launch_bounds__(256) void scatter_kernel(
    const float* __restrict__ h, const int* __restrict__ ei,
    const float* __restrict__ dis, float* __restrict__ acc, int E, int F) {
  int wid = (blockIdx.x * blockDim.x + threadIdx.x) >> 5;
  int nw  = (gridDim.x * blockDim.x) >> 5;
  int lane = threadIdx.x & 31;
  int F2 = F >> 1;
  for (int e = wid; e < E; e += nw) {
    int r = ei[e];
    int c = ei[E + e];
    float s = dis[r];
    const float2* hr = (const float2*)(h + (long long)r * F);
    float* ac = acc + (long long)c * F;
    for (int j = lane; j < F2; j += 32) {
      float2 v = hr[j];
      unsafeAtomicAdd(&ac[2 * j],     s * v.x);
      unsafeAtomicAdd(&ac[2 * j + 1], s * v.y);
    }
  }
}

// h[i] = relu(coef[v]*h[i] + dis[v]*acc[i]),  v = i >> shift
__global__ void combine_relu_kernel(float* __restrict__ h, const float* __restrict__ acc,
                                    const float* __restrict__ dis, const float* __restrict__ coef,
                                    long long total, int shift) {
  long long stride = (long long)gridDim.x * blockDim.x;
  for (long long i = (long long)blockIdx.x * blockDim.x + threadIdx.x; i < total; i += stride) {
    int v = (int)(i >> shift);
    float t = coef[v] * h[i] + dis[v] * acc[i];
    h[i] = fmaxf(t, 0.0f);
  }
}

// combine + log_softmax over feature dim (one wave per node; F<=128)
__global__ __launch_bounds__(256) void final_logsoftmax_kernel(
    float* __restrict__ h2, const float* __restrict__ acc,
    const float* __restrict__ dis, const float* __restrict__ coef, int N, int F) {
  int wid = (blockIdx.x * blockDim.x + threadIdx.x) >> 5;
  int nw  = (gridDim.x * blockDim.x) >> 5;
  int lane = threadIdx.x & 31;
  int F32 = F >> 5;  // per-lane chunks (4 for F=128)
  for (int v = wid; v < N; v += nw) {
    float cf = coef[v], dv = dis[v];
    long long base = (long long)v * F;
    float t[4];
    float mx = -INFINITY;
    for (int j = 0; j < F32; ++j) {
      int f = lane + 32 * j;
      t[j] = cf * h2[base + f] + dv * acc[base + f];
      mx = fmaxf(mx, t[j]);
    }
    for (int o = 16; o > 0; o >>= 1) mx = fmaxf(mx, __shfl_xor(mx, o, 32));
    float s = 0.0f;
    for (int j = 0; j < F32; ++j) s += expf(t[j] - mx);
    for (int o = 16; o > 0; o >>= 1) s += __shfl_xor(s, o, 32);
    float L = logf(s);
    for (int j = 0; j < F32; ++j) h2[base + lane + 32 * j] = t[j] - mx - L;
  }
}

// ---------------------------------------------------------------------------
extern "C" void kernel_launch(void* const* d_in, const int* in_sizes, int n_in,
                              void* d_out, int out_size, void* d_ws, size_t ws_size,
                              hipStream_t stream) {
  const float* x  = (const float*)d_in[0];
  const int*   ei = (const int*)d_in[1];    // [2,E] (harness delivers integers as int32)
  const float* W1 = (const float*)d_in[2];
  const float* b1 = (const float*)d_in[3];
  const float* W2 = (const float*)d_in[4];
  const float* b2 = (const float*)d_in[5];
  float* out = (float*)d_out;

  const int H = in_sizes[3];                 // 64
  const int D = in_sizes[5];                 // 128
  const long long N = in_sizes[0] / D;       // 100000
  const int E = in_sizes[1] / 2;             // 1600000

  // workspace layout (floats):
  //   [0,N)    outdeg u32      [N,2N)   indeg u32
  //   [2N,3N)  dis f32         [3N,4N)  coef f32
  //   [4N, 4N+N*H)        acc1
  //   [4N+N*H, 4N+2*N*H)  h1 / z1
  //   [4N, 4N+N*D)        acc2   (reuses acc1+h1 after GEMM2)
  float* ws = (float*)d_ws;
  unsigned* outdeg = (unsigned*)ws;
  unsigned* indeg  = (unsigned*)(ws + N);
  float* dis  = ws + 2 * N;
  float* coef = ws + 3 * N;
  float* acc1 = ws + 4 * N;
  float* h1   = ws + 4 * N + N * (long long)H;
  float* acc2 = ws + 4 * N;

  int shiftH = 0; while ((1 << shiftH) < H) ++shiftH;

  // 1) zero degree counters
  zero_kernel<<<1024, 256, 0, stream>>>((float4*)ws, (2 * N) / 4);
  // 2) degrees
  degree_kernel<<<2048, 256, 0, stream>>>(ei, E, outdeg, indeg);
  // 3) per-node normalization terms
  node_prep_kernel<<<512, 256, 0, stream>>>(outdeg, indeg, dis, coef, (int)N);

  // 4) GEMM1: h1 = x @ W1^T + b1   [N,H]
  launch_gemm(x, W1, b1, h1, N, D, H, stream);

  // 5) zero acc1, 6) scatter layer-1 messages
  zero_kernel<<<2048, 256, 0, stream>>>((float4*)acc1, N * H / 4);
  scatter_kernel<<<8192, 256, 0, stream>>>(h1, ei, dis, acc1, E, H);
  // 7) z1 = relu(coef*h1 + dis*acc1)  (in place)
  combine_relu_kernel<<<4096, 256, 0, stream>>>(h1, acc1, dis, coef, N * H, shiftH);

  // 8) GEMM2: h2 = z1 @ W2^T + b2  -> d_out  [N,D]
  launch_gemm(h1, W2, b2, out, N, H, D, stream);

  // 9) zero acc2 (safe: h1/acc1 dead after GEMM2), 10) scatter layer-2 messages
  zero_kernel<<<2048, 256, 0, stream>>>((float4*)acc2, N * D / 4);
  scatter_kernel<<<8192, 256, 0, stream>>>(out, ei, dis, acc2, E, D);

  // 11) combine + log_softmax (in place on d_out)
  final_logsoftmax_kernel<<<4096, 256, 0, stream>>>(out, acc2, dis, coef, (int)N, D);
}